// ROIAlignment_72524817760384
// MI455X (gfx1250) — compile-verified
//
#include <hip/hip_runtime.h>
#include <stdint.h>

typedef __attribute__((ext_vector_type(4))) float v4f;

#define CCH 256   // channels per feature map / output

// ---- CDNA5 async-memory path: per-lane 16B gather from global into LDS,
// tracked with ASYNCcnt (emits global_load_async_to_lds_b128 / s_wait_asynccnt).
__device__ __forceinline__ void async_gather_b128(unsigned lds_byte_off, const float* gaddr) {
  asm volatile("global_load_async_to_lds_b128 %0, %1, off"
               :: "v"(lds_byte_off), "v"(gaddr)
               : "memory");
}
__device__ __forceinline__ void wait_async_le4() {
  asm volatile("s_wait_asynccnt 0x4" ::: "memory");   // previous stage (oldest 4) done
}
__device__ __forceinline__ void wait_async_0() {
  asm volatile("s_wait_asynccnt 0x0" ::: "memory");
}

struct Sample {
  const float* tl; const float* tr; const float* bl; const float* br;
  float xl, yl;
  int py, px;
  bool valid;
};

__global__ __launch_bounds__(256, 2) void roi_align_level(
    const float* __restrict__ fm,     // (2, H, H, 256) f32
    const float* __restrict__ boxes,  // (n, 4) f32  [y1,x1,y2,x2]
    const int*   __restrict__ ixes,   // (n, 2) i32  [img, proposal]
    float*       __restrict__ out,    // (2, 2000, 7, 7, 256) f32
    const int H)
{
  // [buf][corner][tid] : 2 * 4 * 256 * 16B = 32 KB (of 320 KB/WGP)
  __shared__ v4f stage[2][4][256];

  const int b    = (int)blockIdx.x;     // one block per box
  const int tid  = (int)threadIdx.x;
  const int cidx = (tid & 63) << 2;     // channel (float4-aligned), 64 lanes cover 256 ch
  const int sg   = tid >> 6;            // sample subgroup 0..3 (4 samples in flight)

  const float by1 = boxes[4*b+0];
  const float bx1 = boxes[4*b+1];
  const float by2 = boxes[4*b+2];
  const float bx2 = boxes[4*b+3];
  const int img  = ixes[2*b+0];
  const int prop = ixes[2*b+1];

  const float hm1   = (float)(H - 1);
  const float ystep = (by2 - by1) * hm1 * (1.0f/6.0f);
  const float xstep = (bx2 - bx1) * hm1 * (1.0f/6.0f);
  const float ybase = by1 * hm1;
  const float xbase = bx1 * hm1;

  auto make_sample = [&](int s) -> Sample {
    Sample si;
    si.py = s / 7;
    si.px = s - si.py * 7;
    const float y = fmaf((float)si.py, ystep, ybase);
    const float x = fmaf((float)si.px, xstep, xbase);
    const bool vy = (y >= 0.0f) && (y <= hm1);
    const bool vx = (x >= 0.0f) && (x <= hm1);
    si.valid = vy && vx;
    const float y0f = floorf(y), x0f = floorf(x);
    si.yl = y - y0f;
    si.xl = x - x0f;
    int y0 = min(max((int)y0f, 0), H - 1);
    int y1 = min(y0 + 1, H - 1);
    int x0 = min(max((int)x0f, 0), H - 1);
    int x1 = min(x0 + 1, H - 1);
    const float* row0 = fm + ((size_t)(img * H + y0) * (size_t)H) * CCH + cidx;
    const float* row1 = fm + ((size_t)(img * H + y1) * (size_t)H) * CCH + cidx;
    si.tl = row0 + (size_t)x0 * CCH;
    si.tr = row0 + (size_t)x1 * CCH;
    si.bl = row1 + (size_t)x0 * CCH;
    si.br = row1 + (size_t)x1 * CCH;
    return si;
  };

  // Raw LDS byte offset of this thread's slot (low 32 bits of the flat address
  // of an addrspace(3) object are the wave-relative LDS offset).
  const unsigned lds0 = (unsigned)(uintptr_t)&stage[0][0][tid];

  auto issue = [&](const Sample& si, int buf) {
    const unsigned base = lds0 + (unsigned)buf * 16384u;   // buf stride
    async_gather_b128(base +     0u, si.tl);               // corner stride 4096B
    async_gather_b128(base +  4096u, si.tr);
    async_gather_b128(base +  8192u, si.bl);
    async_gather_b128(base + 12288u, si.br);
  };

  const int NSTAGE = 13;                 // ceil(49 / 4)
  Sample cur = make_sample(sg);          // stage 0: s = sg (0..3, all valid)
  issue(cur, 0);

  for (int stg = 0; stg < NSTAGE; ++stg) {
    Sample nxt = cur;
    if (stg + 1 < NSTAGE) {
      int s1 = sg + 4 * (stg + 1);
      nxt = make_sample(s1 < 49 ? s1 : 48);   // clamp: harmless loads, store predicated
      issue(nxt, (stg + 1) & 1);
      wait_async_le4();                       // async loads complete in order
    } else {
      wait_async_0();
    }

    const v4f* buf = &stage[stg & 1][0][tid];
    v4f tl = buf[0];
    v4f tr = buf[256];
    v4f bl = buf[512];
    v4f br = buf[768];

    v4f top = tl + (tr - tl) * cur.xl;
    v4f bot = bl + (br - bl) * cur.xl;
    v4f o   = top + (bot - top) * cur.yl;
    if (!cur.valid) {
      v4f z = {0.0f, 0.0f, 0.0f, 0.0f};
      o = z;
    }

    const int s = sg + 4 * stg;
    if (s < 49) {
      // Non-temporal: don't let the 200MB output stream evict the ~178MB of
      // feature maps resident in the 192MB L2.
      size_t off = ((((size_t)img * 2000 + prop) * 7 + cur.py) * 7 + cur.px) * CCH + cidx;
      __builtin_nontemporal_store(o, (v4f*)(out + off));
    }
    cur = nxt;
  }
}

extern "C" void kernel_launch(void* const* d_in, const int* in_sizes, int n_in,
                              void* d_out, int out_size, void* d_ws, size_t ws_size,
                              hipStream_t stream) {
  (void)n_in; (void)out_size; (void)d_ws; (void)ws_size;
  float* out = (float*)d_out;
  const int Hs[4] = {256, 128, 64, 32};
  // setup_inputs() dict order: fm0, boxes0, ixes0, fm1, boxes1, ixes1, ...
  for (int l = 0; l < 4; ++l) {
    const float* fm = (const float*)d_in[3*l + 0];
    const float* bx = (const float*)d_in[3*l + 1];
    const int*   ix = (const int*)d_in[3*l + 2];
    const int nbox  = in_sizes[3*l + 1] / 4;
    if (nbox > 0) {
      roi_align_level<<<dim3((unsigned)nbox), dim3(256), 0, stream>>>(fm, bx, ix, out, Hs[l]);
    }
  }
}